// PACTIntegerSoftmax_33895881900594
// MI455X (gfx1250) — compile-verified
//
#include <hip/hip_runtime.h>

typedef __attribute__((ext_vector_type(2))) float v2f;
typedef __attribute__((ext_vector_type(4))) float v4f;
typedef __attribute__((ext_vector_type(8))) float v8f;

// ---- monotone key mapping so unsigned atomicMax orders like float max ----
__device__ __forceinline__ unsigned f2key(float f) {
    unsigned u = __float_as_uint(f);
    return (u & 0x80000000u) ? ~u : (u | 0x80000000u);
}
__device__ __forceinline__ float key2f(unsigned k) {
    unsigned u = (k & 0x80000000u) ? (k & 0x7FFFFFFFu) : ~k;
    return __uint_as_float(u);
}

__global__ void init_ws_kernel(unsigned* ws) {
    if (threadIdx.x == 0 && blockIdx.x == 0) ws[0] = 0u;  // below every real key
}

// ---- Pass 1: global max over the whole tensor (streaming, NT b128 loads) ----
__global__ __launch_bounds__(256) void global_max_kernel(
    const v4f* __restrict__ x, unsigned* __restrict__ gkey, long n4)
{
    float m = -__builtin_inff();
    long i = (long)blockIdx.x * blockDim.x + threadIdx.x;
    const long stride = (long)gridDim.x * blockDim.x;
    for (; i < n4; i += stride) {
        v4f v = __builtin_nontemporal_load(&x[i]);
        // shaped for v_max3_f32 formation: 2 instrs per 4 elements
        float t = fmaxf(fmaxf(v.x, v.y), v.z);
        m = fmaxf(fmaxf(m, t), v.w);
    }
    // wave32 reduction
    #pragma unroll
    for (int off = 16; off > 0; off >>= 1)
        m = fmaxf(m, __shfl_xor(m, off, 32));
    __shared__ float smax[8];
    const int lane = threadIdx.x & 31, wid = threadIdx.x >> 5;
    if (lane == 0) smax[wid] = m;
    __syncthreads();
    if (threadIdx.x == 0) {
        float bm = smax[0];
        #pragma unroll
        for (int w = 1; w < 8; ++w) bm = fmaxf(bm, smax[w]);
        atomicMax(gkey, f2key(bm));
    }
}

// ---- Pass 2: one wave per 1024-element row; WMMA f32 cross-lane row sum ----
__global__ __launch_bounds__(256) void ibert_softmax_kernel(
    const float* __restrict__ x, float* __restrict__ out,
    const unsigned* __restrict__ gkey,
    const float* __restrict__ cA, const float* __restrict__ cB,
    const float* __restrict__ cC, const float* __restrict__ lg,
    const int* __restrict__ nl)
{
    const int lane = threadIdx.x & 31;
    const int wid  = threadIdx.x >> 5;
    const long row = (long)blockIdx.x * 8 + wid;

    const float gmax  = key2f(*gkey);
    const float a     = *cA;
    const float b     = *cB;
    const float c     = *cC;
    const float l2    = *lg;
    const float invl2 = 1.0f / l2;          // exact for log2 = 4.0
    const float negl2 = -l2;
    const float ab2   = 2.0f * a * b;       // A(p+B)^2+C = (A*p + 2AB)*p + (AB^2+C)
    const float K     = a * b * b + c;
    const float n1    = (float)(*nl - 1);

    const v4f* __restrict__ rx = (const v4f*)(x + row * 1024);
    v4f* __restrict__       ro = (v4f*)(out + row * 1024);

    // Each lane handles 8 float4 chunks: chunk c covers columns c*128 + lane*4.
    float y[32];
    float s0 = 0.f, s1 = 0.f, s2 = 0.f, s3 = 0.f;
    #pragma unroll
    for (int ch = 0; ch < 8; ++ch) {
        v4f v = __builtin_nontemporal_load(&rx[ch * 32 + lane]);
        float vv[4] = {v.x, v.y, v.z, v.w};
        float acc[4];
        #pragma unroll
        for (int j = 0; j < 4; ++j) {
            float xt = vv[j] - gmax;              // <= 0
            float nz = __builtin_ceilf(xt * invl2); // nz = -z = ceil(xt/log2) <= 0
            float p  = fmaf(nz, negl2, xt);       // p = xt + z*log2
            float g  = fmaf(a, p, ab2);
            float y0 = fmaf(p, g, K);             // A(p+B)^2 + C
            float yy = ldexpf(y0, (int)nz);       // * 2^-z
            y[ch * 4 + j] = yy;
            acc[j] = yy;
        }
        s0 += acc[0]; s1 += acc[1]; s2 += acc[2]; s3 += acc[3];
    }
    float s = (s0 + s1) + (s2 + s3);

    // Cross-lane reduction via V_WMMA_F32_16X16X4_F32 (exact fp32):
    //   A[m,0]=s_m, A[m,2]=s_{m+16}, rest 0 ; B = ones(4x16)
    //   => D[m,n] = s_m + s_{m+16} for all n.
    v2f A; A.x = s;    A.y = 0.0f;
    v2f B; B.x = 1.0f; B.y = 1.0f;
    v8f C = {};
    C = __builtin_amdgcn_wmma_f32_16x16x4_f32(
            /*neg_a=*/false, A, /*neg_b=*/false, B,
            /*c_mod=*/(short)0, C, /*reuse_a=*/false, /*reuse_b=*/false);
    // Lanes 0-15 hold rows 0..7, lanes 16-31 hold rows 8..15: sum my 8, then
    // one SWAPX16 ds_swizzle pairs the half-waves for the full 32-lane sum.
    float half = ((C[0] + C[1]) + (C[2] + C[3])) + ((C[4] + C[5]) + (C[6] + C[7]));
    float other = __int_as_float(
        __builtin_amdgcn_ds_swizzle(__float_as_int(half), 0x401F)); // xor 0x10
    float ysum = half + other;

    const float r = n1 / ysum;  // one divide per thread

    #pragma unroll
    for (int ch = 0; ch < 8; ++ch) {
        v4f o;
        o.x = __builtin_floorf(y[ch * 4 + 0] * r);
        o.y = __builtin_floorf(y[ch * 4 + 1] * r);
        o.z = __builtin_floorf(y[ch * 4 + 2] * r);
        o.w = __builtin_floorf(y[ch * 4 + 3] * r);
        __builtin_nontemporal_store(o, &ro[ch * 32 + lane]);
    }
}

extern "C" void kernel_launch(void* const* d_in, const int* in_sizes, int n_in,
                              void* d_out, int out_size, void* d_ws, size_t ws_size,
                              hipStream_t stream) {
    const float* x  = (const float*)d_in[0];
    const float* cA = (const float*)d_in[1];
    const float* cB = (const float*)d_in[2];
    const float* cC = (const float*)d_in[3];
    const float* lg = (const float*)d_in[4];
    const int*   nl = (const int*)d_in[5];
    float* out = (float*)d_out;
    unsigned* gkey = (unsigned*)d_ws;

    const long n    = (long)in_sizes[0];   // 100,663,296
    const long n4   = n / 4;
    const long rows = n / 1024;            // 98,304

    init_ws_kernel<<<1, 64, 0, stream>>>(gkey);

    int maxBlocks = 12288;
    global_max_kernel<<<maxBlocks, 256, 0, stream>>>((const v4f*)x, gkey, n4);

    int smBlocks = (int)(rows / 8);        // 8 waves/block, 1 wave per row
    ibert_softmax_kernel<<<smBlocks, 256, 0, stream>>>(x, out, gkey, cA, cB, cC, lg, nl);
}